// Restore_60979945668667
// MI455X (gfx1250) — compile-verified
//
#include <hip/hip_runtime.h>

// Problem constants (match the reference).
#define NUM_CLUSTER 8
#define ENC_IN      128
#define PER         16          // ENC_IN / NUM_CLUSTER
#define LSEQ        96
#define DMODEL      512
#define BATCH       4

typedef __attribute__((ext_vector_type(4))) float v4f;
typedef __attribute__((ext_vector_type(4))) int   i4;
// Builtin parameter types: int4 pointers in global (AS1) / LDS (AS3).
typedef __attribute__((address_space(1))) i4 gi4;
typedef __attribute__((address_space(3))) i4 li4;

constexpr int ROW_FLOATS     = LSEQ * DMODEL;           // 49152 floats per (b,i) row
constexpr int ROW_F4         = ROW_FLOATS / 4;          // 12288 float4 per row
constexpr int TPB            = 256;                     // 8 waves (wave32)
constexpr int F4_PER_THREAD  = 8;
constexpr int F4_PER_BLOCK   = TPB * F4_PER_THREAD;     // 2048 float4 = 32 KB
constexpr int BLOCKS_PER_ROW = ROW_F4 / F4_PER_BLOCK;   // 6
constexpr int NROWS_OUT      = BATCH * ENC_IN;          // 512

#if __has_builtin(__builtin_amdgcn_global_load_async_to_lds_b128) && \
    __has_builtin(__builtin_amdgcn_global_store_async_from_lds_b128) && \
    __has_builtin(__builtin_amdgcn_s_wait_asynccnt)
#define USE_ASYNC_LDS 1
#else
#define USE_ASYNC_LDS 0
#endif

// ---------------------------------------------------------------------------
// Kernel A (defined first so it leads the assembly): streaming permutation
// copy at 128-bit granularity.
//   out[b,i,:,:] = x[y_pred[i], b, rank[i], :, :]   (contiguous 192 KB rows)
// Each block moves a 32 KB chunk of one row, staged through LDS with the
// CDNA5 async copy engine (ASYNCcnt-tracked, no VGPR round-trip).  Each
// thread owns disjoint LDS slots stage[k*TPB+tid], so per-wave
// s_wait_asynccnt 0 is the only synchronization required.
// ---------------------------------------------------------------------------
__global__ __launch_bounds__(TPB) void gather_copy_kernel(
    const float* __restrict__ x, const int* __restrict__ srcRow0,
    float* __restrict__ out) {
  const int blk   = blockIdx.x;
  const int chunk = blk % BLOCKS_PER_ROW;
  const int row   = blk / BLOCKS_PER_ROW;     // row = b*ENC_IN + i
  const int b     = row >> 7;
  const int i     = row & (ENC_IN - 1);
  const int tid   = threadIdx.x;

  const int srow = srcRow0[i] + b * PER;      // source row index in x

  const v4f* __restrict__ src =
      (const v4f*)x + (size_t)srow * ROW_F4 + (size_t)chunk * F4_PER_BLOCK + tid;
  v4f* __restrict__ dst =
      (v4f*)out + (size_t)row * ROW_F4 + (size_t)chunk * F4_PER_BLOCK + tid;

#if USE_ASYNC_LDS
  __shared__ v4f stage[F4_PER_BLOCK];         // 32 KB staging buffer

#pragma unroll
  for (int k = 0; k < F4_PER_THREAD; ++k) {
    __builtin_amdgcn_global_load_async_to_lds_b128(
        (gi4*)(void*)(src + k * TPB), (li4*)&stage[k * TPB + tid], 0, 0);
  }
  __builtin_amdgcn_s_wait_asynccnt(0);   // this wave's LDS writes landed
#pragma unroll
  for (int k = 0; k < F4_PER_THREAD; ++k) {
    __builtin_amdgcn_global_store_async_from_lds_b128(
        (gi4*)(void*)(dst + k * TPB), (li4*)&stage[k * TPB + tid], 0, 0);
  }
  __builtin_amdgcn_s_wait_asynccnt(0);   // stores drained before wave end
#else
  // Fallback: plain nontemporal 128-bit copies (still bandwidth-optimal).
  v4f v[F4_PER_THREAD];
#pragma unroll
  for (int k = 0; k < F4_PER_THREAD; ++k)
    v[k] = __builtin_nontemporal_load(src + k * TPB);
#pragma unroll
  for (int k = 0; k < F4_PER_THREAD; ++k)
    __builtin_nontemporal_store(v[k], dst + k * TPB);
#endif
}

// ---------------------------------------------------------------------------
// Kernel B: compute source row base for each of the 128 gather indices.
// srcRow0[i] = y_pred[i] * (BATCH*PER) + rank[i], where rank[i] is the number
// of j < i with y_pred[j] == y_pred[i].  1 block x 128 threads; keep codegen
// scalar — this kernel is noise next to the 200 MB copy.
// ---------------------------------------------------------------------------
__global__ __launch_bounds__(ENC_IN) void build_index_kernel(
    const int* __restrict__ y_pred, int* __restrict__ srcRow0) {
  const int i = threadIdx.x;
  const int c = y_pred[i];
  int r = 0;
#pragma clang loop vectorize(disable) unroll(disable)
  for (int j = 0; j < i; ++j) r += (y_pred[j] == c) ? 1 : 0;
  srcRow0[i] = c * (BATCH * PER) + r;
}

// ---------------------------------------------------------------------------
extern "C" void kernel_launch(void* const* d_in, const int* in_sizes, int n_in,
                              void* d_out, int out_size, void* d_ws,
                              size_t ws_size, hipStream_t stream) {
  const float* x      = (const float*)d_in[0];   // (8,4,16,96,512) fp32
  const int*   y_pred = (const int*)d_in[1];     // (128,) int32
  // d_in[2] is the scalar B (=4); compile-time constant here.

  int* srcRow0 = (int*)d_ws;                     // 128 ints of scratch

  build_index_kernel<<<1, ENC_IN, 0, stream>>>(y_pred, srcRow0);
  gather_copy_kernel<<<NROWS_OUT * BLOCKS_PER_ROW, TPB, 0, stream>>>(
      x, srcRow0, (float*)d_out);
}